// QuantizedEmbedding_90503550861770
// MI455X (gfx1250) — compile-verified
//
#include <hip/hip_runtime.h>
#include <hip/hip_bf16.h>

// MI455X plan:
//   readout GEMV streams the full 1.05GB fp32 weight table -> ~45us HBM floor;
//   fuse quantize (rintf + med3 clamp) inline, accumulate via V_WMMA_F32_16X16X4_F32
//   (xq broadcast across B's 16 columns so D column 0 = GEMV result).
//   embed gather (134MB traffic) and xq prep are cheap side kernels.

typedef __attribute__((ext_vector_type(2))) float v2f;
typedef __attribute__((ext_vector_type(8))) float v8f;

#define VOCAB 128000
#define DIM   2048

__device__ __forceinline__ float quantw(float w) {
    // clip(round(w), -128, 127); round-half-even == jnp.round
    return __builtin_amdgcn_fmed3f(rintf(w), -128.0f, 127.0f);
}

// ---------------------------------------------------------------------------
// Kernel 1: dynamic absmax int8 fake-quantization of x -> xq (in workspace)
// ---------------------------------------------------------------------------
__global__ void quantize_x_kernel(const float* __restrict__ x,
                                  float* __restrict__ xq) {
    __shared__ float red[256];
    float m = 0.0f;
    for (int i = threadIdx.x; i < DIM; i += 256)
        m = fmaxf(m, fabsf(x[i]));
    red[threadIdx.x] = m;
    __syncthreads();
    for (int s = 128; s > 0; s >>= 1) {
        if (threadIdx.x < s)
            red[threadIdx.x] = fmaxf(red[threadIdx.x], red[threadIdx.x + s]);
        __syncthreads();
    }
    float scale = fmaxf(red[0] / 127.0f, 1e-12f);
    for (int i = threadIdx.x; i < DIM; i += 256) {
        float q = __builtin_amdgcn_fmed3f(rintf(x[i] / scale), -128.0f, 127.0f);
        xq[i] = q * scale;
    }
}

// ---------------------------------------------------------------------------
// Kernel 2: embedding gather: out[t,:] = quant(W[tok,:]) * scales[tok]
// one block per token; 256 threads x 2 float4 = 2048 floats
// ---------------------------------------------------------------------------
__global__ void embed_kernel(const float* __restrict__ W,
                             const float* __restrict__ scales,
                             const int*   __restrict__ tokens,
                             float*       __restrict__ out) {
    const int t   = blockIdx.x;
    const int tok = tokens[t];
    const float s = scales[tok];
    const float4* __restrict__ wrow = (const float4*)(W + (size_t)tok * DIM);
    float4* __restrict__ orow       = (float4*)(out + (size_t)t * DIM);
    for (int i = threadIdx.x; i < DIM / 4; i += 256) {
        float4 v = wrow[i];
        v.x = quantw(v.x) * s;
        v.y = quantw(v.y) * s;
        v.z = quantw(v.z) * s;
        v.w = quantw(v.w) * s;
        orow[i] = v;
    }
}

// ---------------------------------------------------------------------------
// Kernel 3: readout GEMV via WMMA f32 16x16x4.
//   wave -> 16 rows. A tile: lane l holds row (l&15), K pair (+0,+1 | +2,+3)
//   -> single b64 load/lane. B tile: xq broadcast over all 16 columns
//   -> single ds_load_b64/lane from LDS. D column N=0 (lanes 0 & 16) holds
//   the dot products; scale by scales[row] at store.
// ---------------------------------------------------------------------------
__global__ void readout_wmma_kernel(const float* __restrict__ W,
                                    const float* __restrict__ scales,
                                    const float* __restrict__ xq,
                                    float*       __restrict__ out) {
    __shared__ float lxq[DIM];
    for (int i = threadIdx.x; i < DIM; i += 256)
        lxq[i] = xq[i];
    __syncthreads();

    const int wave = threadIdx.x >> 5;          // 0..7
    const int lane = threadIdx.x & 31;          // 0..31
    const int rowBase = blockIdx.x * 128 + wave * 16;
    const int m    = lane & 15;                 // row within tile
    const int koff = (lane >> 4) << 1;          // 0 (lanes 0-15) or 2 (lanes 16-31)

    const float* __restrict__ wrow = W + (size_t)(rowBase + m) * DIM + koff;
    const float* __restrict__ bsrc = lxq + koff;

    v8f c = {};
#pragma unroll 4
    for (int k = 0; k < DIM; k += 4) {
        // A operand: two consecutive quantized weights of this lane's row
        v2f aw = *(const v2f*)(wrow + k);
        v2f a;
        a.x = quantw(aw.x);
        a.y = quantw(aw.y);
        // B operand: xq[k+koff], xq[k+koff+1] (broadcast across N within lane half)
        v2f b = *(const v2f*)(bsrc + k);
        // D = A(16x4) * B(4x16) + C ; args: neg_a, A, neg_b, B, c_mod, C, reuse_a, reuse_b
        c = __builtin_amdgcn_wmma_f32_16x16x4_f32(false, a, false, b,
                                                  (short)0, c, false, false);
    }

    // Column N=0 of D: lane 0 holds M=0..7 in c[0..7], lane 16 holds M=8..15.
    if ((lane & 15) == 0) {
        const int mb = rowBase + ((lane >> 4) << 3);
#pragma unroll
        for (int v = 0; v < 8; ++v)
            out[mb + v] = c[v] * scales[mb + v];
    }
}

extern "C" void kernel_launch(void* const* d_in, const int* in_sizes, int n_in,
                              void* d_out, int out_size, void* d_ws, size_t ws_size,
                              hipStream_t stream) {
    const float* W      = (const float*)d_in[0];   // [VOCAB, DIM]
    const float* scales = (const float*)d_in[1];   // [VOCAB]
    const int*   tokens = (const int*)d_in[2];     // [8192]
    const float* x      = (const float*)d_in[3];   // [DIM]

    const int n_tokens = in_sizes[2];              // 8192

    float* embed_out   = (float*)d_out;                              // [8192, DIM]
    float* readout_out = (float*)d_out + (size_t)n_tokens * DIM;     // [VOCAB]
    float* xq          = (float*)d_ws;                               // [DIM]

    quantize_x_kernel<<<1, 256, 0, stream>>>(x, xq);
    embed_kernel<<<n_tokens, 256, 0, stream>>>(W, scales, tokens, embed_out);
    readout_wmma_kernel<<<VOCAB / 128, 256, 0, stream>>>(W, scales, xq, readout_out);
}